// MultiBoxLoss_47837345743400
// MI455X (gfx1250) — compile-verified
//
#include <hip/hip_runtime.h>
#include <math.h>

#define BB 32
#define PP 32768
#define CC 81
#define NOBJ 16

typedef float v2f  __attribute__((ext_vector_type(2)));
typedef float v8f  __attribute__((ext_vector_type(8)));
typedef unsigned u32x4 __attribute__((ext_vector_type(4)));
typedef int   i32x4 __attribute__((ext_vector_type(4)));
typedef int   i32x8 __attribute__((ext_vector_type(8)));

#if __has_builtin(__builtin_amdgcn_tensor_load_to_lds) && __has_builtin(__builtin_amdgcn_s_wait_tensorcnt)
#define USE_TDM 1
#else
#define USE_TDM 0
#endif

__device__ __forceinline__ float smooth_l1(float d) {
    float a = fabsf(d);
    return (a < 1.0f) ? 0.5f * a * a : a - 0.5f;
}

__device__ __forceinline__ unsigned rfl(unsigned x) {
    return (unsigned)__builtin_amdgcn_readfirstlane((int)x);
}

// ---------------------------------------------------------------------------
// Kernel 0: zero the small accumulator block (harness poisons ws with 0xAA).
// ---------------------------------------------------------------------------
__global__ void mbox_init(float* accum, int* npos) {
    int t = threadIdx.x;
    if (t < 4) accum[t] = 0.0f;
    if (t >= 4 && t < 4 + BB) npos[t - 4] = 0;
}

// ---------------------------------------------------------------------------
// Kernel 1: per-(row, GT) argmax of IoU over all P priors (first-max index,
// matching jnp.argmax tie-breaking). One block per batch row.
// ---------------------------------------------------------------------------
__global__ __launch_bounds__(256) void mbox_bestprior(
    const float* __restrict__ priors, const float* __restrict__ tboxes,
    int* __restrict__ bestPrior)
{
    __shared__ float gtb[NOBJ * 4];
    __shared__ float gtArea[NOBJ];
    __shared__ float sV[NOBJ][256];
    __shared__ int   sI[NOBJ][256];

    const int b = blockIdx.x, tid = threadIdx.x;
    if (tid < NOBJ * 4) gtb[tid] = tboxes[(size_t)b * NOBJ * 4 + tid];
    __syncthreads();
    if (tid < NOBJ)
        gtArea[tid] = (gtb[tid*4+2] - gtb[tid*4+0]) * (gtb[tid*4+3] - gtb[tid*4+1]);
    __syncthreads();

    float bv[NOBJ];
    int   bi[NOBJ];
#pragma unroll
    for (int j = 0; j < NOBJ; ++j) { bv[j] = -1.0f; bi[j] = 0; }

    for (int p = tid; p < PP; p += 256) {
        float pcx = priors[p*4+0], pcy = priors[p*4+1];
        float pw  = priors[p*4+2], ph  = priors[p*4+3];
        float px1 = pcx - 0.5f*pw, py1 = pcy - 0.5f*ph;
        float px2 = pcx + 0.5f*pw, py2 = pcy + 0.5f*ph;
        float areaP = pw * ph;
#pragma unroll
        for (int j = 0; j < NOBJ; ++j) {
            float iw = fminf(px2, gtb[j*4+2]) - fmaxf(px1, gtb[j*4+0]);
            float ih = fminf(py2, gtb[j*4+3]) - fmaxf(py1, gtb[j*4+1]);
            iw = fmaxf(iw, 0.0f); ih = fmaxf(ih, 0.0f);
            float inter = iw * ih;
            float iou = inter / (areaP + gtArea[j] - inter);
            if (iou > bv[j]) { bv[j] = iou; bi[j] = p; }  // strict > keeps first
        }
    }
#pragma unroll
    for (int j = 0; j < NOBJ; ++j) { sV[j][tid] = bv[j]; sI[j][tid] = bi[j]; }
    __syncthreads();
    if (tid < NOBJ) {
        float best = -1.0f; int bidx = 0x7fffffff;
        for (int t = 0; t < 256; ++t) {
            float v = sV[tid][t]; int i = sI[tid][t];
            if (v > best || (v == best && i < bidx)) { best = v; bidx = i; }
        }
        bestPrior[b * NOBJ + tid] = bidx;
    }
}

// ---------------------------------------------------------------------------
// Kernel 2 (heavy, memory-bound): each wave stages its 16x81 f32 conf tile
// into LDS with one TDM tensor_load_to_lds (contiguous 5184B 2D tile), waits
// on TENSORcnt, then computes per-prior logsumexp via chained
// V_WMMA_F32_16X16X4_F32 (A = exp tile, B = ones), plus match recompute,
// smooth-L1, neg-loss array and global accumulators.
// ---------------------------------------------------------------------------
__global__ __launch_bounds__(256) void mbox_main(
    const float* __restrict__ pred_loc, const float* __restrict__ pred_conf,
    const float* __restrict__ priors,   const int*   __restrict__ labels,
    const float* __restrict__ tboxes,   const int*   __restrict__ bestPrior,
    float* __restrict__ negLoss, float* __restrict__ accum, int* __restrict__ npos)
{
    __shared__ __align__(16) float tiles[8][16 * CC];   // 41472 B, one tile/wave
    __shared__ float gtb[NOBJ * 4];
    __shared__ float gtArea[NOBJ];
    __shared__ int   gbp[NOBJ];
    __shared__ int   glab[NOBJ];
    __shared__ float sL[8], sC[8];
    __shared__ int   sN[8];

    const int tid  = threadIdx.x;
    const int wave = tid >> 5, lane = tid & 31;
    const int tile0 = blockIdx.x * 8;          // 8 tiles per block, same row b
    const int b = tile0 >> 11;                 // 2048 tiles per row

    if (tid < NOBJ * 4) gtb[tid] = tboxes[(size_t)b * NOBJ * 4 + tid];
    if (tid < NOBJ) { gbp[tid] = bestPrior[b*NOBJ + tid]; glab[tid] = labels[b*NOBJ + tid]; }
    __syncthreads();
    if (tid < NOBJ)
        gtArea[tid] = (gtb[tid*4+2] - gtb[tid*4+0]) * (gtb[tid*4+3] - gtb[tid*4+1]);
    __syncthreads();

    const int tile  = tile0 + wave;
    const int pbase = (tile & 2047) << 4;
    const int m  = lane & 15;                  // matrix row == prior within tile
    const int hk = lane >> 4;                  // K half: lanes16-31 hold K2,K3
    const int p  = pbase + m;
    float* tp = &tiles[wave][0];
    const float* gsrc = pred_conf + ((size_t)b * PP + pbase) * CC;

#if USE_TDM
    {
        // ---- Tensor Data Mover: 2D tile, 16 rows x 81 f32, stride 81 -------
        unsigned long long ga = (unsigned long long)(uintptr_t)gsrc;
        unsigned lds_off = (unsigned)(uintptr_t)tp;     // low 32 = LDS offset
        u32x4 g0 = { rfl(1u),                           // count=1, user mode
                     rfl(lds_off),                      // lds_addr [63:32]
                     rfl((unsigned)ga),                 // global_addr lo
                     rfl(((unsigned)(ga >> 32) & 0x1FFFFFFu) | 0x80000000u) }; // ga hi | type=2
        i32x8 g1 = { (int)rfl(0x00020000u),             // data_size=4B, wg_mask=0
                     (int)rfl((unsigned)(CC << 16)),    // tensor_dim0 = 81
                     (int)rfl((unsigned)(16 << 16)),    // tensor_dim1 = 16
                     (int)rfl((unsigned)(CC << 16)),    // tile_dim0 = 81
                     (int)rfl(16u),                     // tile_dim1 = 16
                     (int)rfl((unsigned)CC),            // tensor_dim0_stride = 81
                     0, 0 };
        i32x4 z4 = { 0, 0, 0, 0 };
        i32x8 z8 = { 0, 0, 0, 0, 0, 0, 0, 0 };
        // 6-arg form (amdgpu-toolchain / clang-23): (g0, g1, g2, g3, g4, cpol)
        __builtin_amdgcn_tensor_load_to_lds(g0, g1, z4, z4, z8, 0);
        __builtin_amdgcn_s_wait_tensorcnt(0);
    }
#else
    {
        // Fallback: coalesced cooperative copy (tile base is 16B aligned).
        const float4* g4 = (const float4*)gsrc;
        float4* t4 = (float4*)tp;
        for (int i = lane; i < (16 * CC) / 4; i += 32) t4[i] = g4[i];
        __syncthreads();
    }
#endif

    // ---- pass 1: per-prior max over 81 classes (from LDS) ------------------
    const float* rowp = tp + m * CC;
    float mx = -3.402823466e38f;
#pragma unroll
    for (int ch = 0; ch < 20; ++ch) {
        float a0 = rowp[ch * 4 + hk * 2 + 0];
        float a1 = rowp[ch * 4 + hk * 2 + 1];
        mx = fmaxf(mx, fmaxf(a0, a1));
    }
    if (hk == 0) mx = fmaxf(mx, rowp[80]);
    mx = fmaxf(mx, __shfl_xor(mx, 16, 32));    // combine K halves

    // ---- pass 2: sum(exp(x-mx)) via WMMA: D = A(16x4) * ones(4x16) + C -----
    v8f acc = {0.f, 0.f, 0.f, 0.f, 0.f, 0.f, 0.f, 0.f};
    const v2f ones = {1.0f, 1.0f};
#pragma unroll
    for (int ch = 0; ch < 20; ++ch) {
        v2f a;
        a.x = __expf(rowp[ch * 4 + hk * 2 + 0] - mx);   // native v_exp_f32
        a.y = __expf(rowp[ch * 4 + hk * 2 + 1] - mx);
        acc = __builtin_amdgcn_wmma_f32_16x16x4_f32(false, a, false, ones,
                                                    (short)0, acc, false, false);
    }
    {   // tail chunk: only class 80 valid (lower K half); pad contributes 0
        v2f a;
        a.x = (hk == 0) ? __expf(rowp[80] - mx) : 0.0f;
        a.y = 0.0f;
        acc = __builtin_amdgcn_wmma_f32_16x16x4_f32(false, a, false, ones,
                                                    (short)0, acc, false, false);
    }
    // D layout: VGPR r = row r (lanes 0-15) / row r+8 (lanes 16-31), row sum
    // replicated across columns. Route row m's sum to lanes m, m+16.
    float sel = acc[lane & 7];
    int   src = (m < 8) ? m : (m + 8);
    float rowsum = __shfl(sel, src, 32);
    float lse = mx + __logf(rowsum);

    // ---- per-prior match / losses (lanes 0-15 carry the 16 priors) ---------
    float cL = 0.0f, cC = 0.0f; int cN = 0;
    if (lane < 16) {
        float pcx = priors[p*4+0], pcy = priors[p*4+1];
        float pw  = priors[p*4+2], ph  = priors[p*4+3];
        float px1 = pcx - 0.5f*pw, py1 = pcy - 0.5f*ph;
        float px2 = pcx + 0.5f*pw, py2 = pcy + 0.5f*ph;
        float areaP = pw * ph;
        float bov = -1.0f; int bj = 0;
#pragma unroll
        for (int j = 0; j < NOBJ; ++j) {
            float iw = fminf(px2, gtb[j*4+2]) - fmaxf(px1, gtb[j*4+0]);
            float ih = fminf(py2, gtb[j*4+3]) - fmaxf(py1, gtb[j*4+1]);
            iw = fmaxf(iw, 0.0f); ih = fmaxf(ih, 0.0f);
            float inter = iw * ih;
            float iou = inter / (areaP + gtArea[j] - inter);
            if (iou > bov) { bov = iou; bj = j; }   // first-max like argmax
        }
#pragma unroll
        for (int j = 0; j < NOBJ; ++j)              // force-match override
            if (gbp[j] == p) { bj = j; bov = 2.0f; }

        int  conf_t = (bov < 0.5f) ? 0 : (glab[bj] + 1);
        bool pos = (conf_t != 0);
        float lossc = lse - rowp[conf_t];           // gather from LDS tile
        negLoss[(size_t)b * PP + p] = pos ? 0.0f : lossc;

        if (pos) {
            cC = lossc; cN = 1;
            float tx1 = gtb[bj*4+0], ty1 = gtb[bj*4+1];
            float tx2 = gtb[bj*4+2], ty2 = gtb[bj*4+3];
            float gcx = ((tx1 + tx2) * 0.5f - pcx) / (0.1f * pw);
            float gcy = ((ty1 + ty2) * 0.5f - pcy) / (0.1f * ph);
            float gw  = logf((tx2 - tx1) / pw) * 5.0f;   // /0.2 (cold path)
            float gh  = logf((ty2 - ty1) / ph) * 5.0f;
            const float* pl = pred_loc + ((size_t)b * PP + p) * 4;
            cL = smooth_l1(pl[0] - gcx) + smooth_l1(pl[1] - gcy) +
                 smooth_l1(pl[2] - gw)  + smooth_l1(pl[3] - gh);
        }
    }

    // ---- block reduction + global accumulate -------------------------------
#pragma unroll
    for (int off = 16; off > 0; off >>= 1) {
        cL += __shfl_down(cL, off, 32);
        cC += __shfl_down(cC, off, 32);
        cN += __shfl_down(cN, off, 32);
    }
    if (lane == 0) { sL[wave] = cL; sC[wave] = cC; sN[wave] = cN; }
    __syncthreads();
    if (tid == 0) {
        float tL = 0.0f, tC = 0.0f; int tN = 0;
        for (int w = 0; w < 8; ++w) { tL += sL[w]; tC += sC[w]; tN += sN[w]; }
        atomicAdd(&accum[0], tL);
        atomicAdd(&accum[1], tC);
        atomicAdd(&npos[b], tN);
    }
}

// ---------------------------------------------------------------------------
// Kernel 3: per-row exact top-k sum via 4-round radix select on float bits
// (all losses >= 0 so uint order == float order). One block per row.
// ---------------------------------------------------------------------------
__global__ __launch_bounds__(256) void mbox_topk(
    const float* __restrict__ negLoss, const int* __restrict__ npos,
    float* __restrict__ accum)
{
    __shared__ unsigned hist[256];
    __shared__ unsigned s_chosen;
    __shared__ int      s_need;
    __shared__ float    red[256];

    const int b = blockIdx.x, tid = threadIdx.x;
    const float* row = negLoss + (size_t)b * PP;
    int k = 3 * npos[b];
    if (k > PP - 1) k = PP - 1;
    if (k <= 0) return;                         // block-uniform: no barriers hit

    unsigned prefix = 0;
    int need = k;
#pragma unroll
    for (int shift = 24; shift >= 0; shift -= 8) {
        hist[tid] = 0;
        __syncthreads();
        unsigned hmask = (shift == 24) ? 0u : (0xFFFFFFFFu << (shift + 8));
        for (int p = tid; p < PP; p += 256) {
            unsigned key = __float_as_uint(row[p]);
            if ((key & hmask) == (prefix & hmask))
                atomicAdd(&hist[(key >> shift) & 255], 1u);
        }
        __syncthreads();
        if (tid == 0) {
            unsigned cum = 0; int bsel = 0;
            for (int bb = 255; bb >= 0; --bb) {
                unsigned h = hist[bb];
                if (cum + h >= (unsigned)need) { bsel = bb; break; }
                cum += h;
            }
            s_chosen = (unsigned)bsel;
            s_need = need - (int)cum;
        }
        __syncthreads();
        prefix |= (s_chosen << shift);
        need = s_need;
    }
    // prefix = bit pattern of the k-th largest; include `need` copies of it.
    float thr = __uint_as_float(prefix);
    float partial = 0.0f;
    for (int p = tid; p < PP; p += 256) {
        float v = row[p];
        if (__float_as_uint(v) > prefix) partial += v;
    }
    red[tid] = partial;
    __syncthreads();
#pragma unroll
    for (int s = 128; s > 0; s >>= 1) {
        if (tid < s) red[tid] += red[tid + s];
        __syncthreads();
    }
    if (tid == 0) atomicAdd(&accum[2], red[0] + (float)need * thr);
}

// ---------------------------------------------------------------------------
// Kernel 4: finalize the two scalars.
// ---------------------------------------------------------------------------
__global__ void mbox_final(const float* accum, const int* npos, float* out) {
    int N = 0;
    for (int b = 0; b < BB; ++b) N += npos[b];
    float fn = (float)N;
    out[0] = accum[0] / fn;
    out[1] = (accum[1] + accum[2]) / fn;
}

// ---------------------------------------------------------------------------
extern "C" void kernel_launch(void* const* d_in, const int* in_sizes, int n_in,
                              void* d_out, int out_size, void* d_ws, size_t ws_size,
                              hipStream_t stream) {
    const float* pred_loc  = (const float*)d_in[0];
    const float* pred_conf = (const float*)d_in[1];
    const float* priors    = (const float*)d_in[2];
    const int*   labels    = (const int*)d_in[3];
    const float* tboxes    = (const float*)d_in[4];
    float* out = (float*)d_out;

    // Workspace layout: [negLoss B*P floats][bestPrior B*16 ints][accum 4 f][npos 32 i]
    float* negLoss   = (float*)d_ws;
    int*   bestPrior = (int*)((char*)d_ws + (size_t)BB * PP * sizeof(float));
    float* accum     = (float*)((char*)bestPrior + BB * NOBJ * sizeof(int));
    int*   npos      = (int*)(accum + 4);

    mbox_init<<<1, 64, 0, stream>>>(accum, npos);
    mbox_bestprior<<<BB, 256, 0, stream>>>(priors, tboxes, bestPrior);
    mbox_main<<<BB * (PP / 16) / 8, 256, 0, stream>>>(
        pred_loc, pred_conf, priors, labels, tboxes, bestPrior,
        negLoss, accum, npos);
    mbox_topk<<<BB, 256, 0, stream>>>(negLoss, npos, accum);
    mbox_final<<<1, 1, 0, stream>>>(accum, npos, out);
    (void)in_sizes; (void)n_in; (void)out_size; (void)ws_size;
}